// AIGGCN_48576080117931
// MI455X (gfx1250) — compile-verified
//
#include <hip/hip_runtime.h>

typedef __attribute__((ext_vector_type(2))) float v2f;
typedef __attribute__((ext_vector_type(8))) float v8f;

#define NNODES 50000
#define NEDGES 800000
#define DDIM   256
#define MTILES (NNODES / 16)   // 3125, exact

// ---------------------------------------------------------------- edge MLP
__global__ void k_edge_mlp(const float* __restrict__ ea,
                           const float* __restrict__ mw1, const float* __restrict__ mb1,
                           const float* __restrict__ mw2, const float* __restrict__ mb2,
                           float* __restrict__ ew) {
    int e = blockIdx.x * blockDim.x + threadIdx.x;
    if (e >= NEDGES) return;
    float a = ea[e];
    float s = mb2[0];
#pragma unroll
    for (int j = 0; j < 8; ++j) {
        float h = fmaxf(a * mw1[j] + mb1[j], 0.0f);
        s += h * mw2[j];
    }
    ew[e] = 1.0f / (1.0f + expf(-s));
}

// ---------------------------------------------------------------- degree
__global__ void k_deg_init(float* __restrict__ deg) {
    int i = blockIdx.x * blockDim.x + threadIdx.x;
    if (i < NNODES) deg[i] = 1.0f;   // self-loop contributes 1
}

__global__ void k_deg_acc(const int* __restrict__ dst, const float* __restrict__ ew,
                          float* __restrict__ deg) {
    int e = blockIdx.x * blockDim.x + threadIdx.x;
    if (e >= NEDGES) return;
    __hip_atomic_fetch_add(&deg[dst[e]], ew[e], __ATOMIC_RELAXED, __HIP_MEMORY_SCOPE_AGENT);
}

__global__ void k_dinv(const float* __restrict__ deg, float* __restrict__ dinv,
                       float* __restrict__ invdeg) {
    int i = blockIdx.x * blockDim.x + threadIdx.x;
    if (i >= NNODES) return;
    float d = deg[i];
    dinv[i]   = rsqrtf(d);
    invdeg[i] = 1.0f / d;
}

// ---------------------------------------------------------------- GEMM: H = act(A) @ W + b
// block = 256 threads (8 waves). Block tile: 128 rows x 64 cols. Full K=256 of the
// 64-col W slab staged in 64 KB LDS (K-pair-interleaved layout), shared by all 8
// waves. Each wave: 16 rows, four 16x16 f32 accumulators, K stepped by 4 via
// V_WMMA_F32_16X16X4_F32.
//
// LDS layout: element (k, n_local) at word (k>>2)*256 + n_local*4 + (k&3).
//  -> a lane's B fragment (W[k+kb][n], W[k+kb+1][n]) is one aligned b64,
//     conflict-free across the wave (16 n-values x 4 words = 64 banks).
template <bool RELU>
__global__ __launch_bounds__(256) void k_gemm(const float* __restrict__ A,
                                              const float* __restrict__ W,
                                              const float* __restrict__ bias,
                                              float* __restrict__ H) {
    __shared__ float sW[DDIM * 64];        // 64 KB
    const int tid = threadIdx.x;
    const int n0  = blockIdx.y * 64;

    // cooperative fill: each thread writes one float4 = K-quad of one column.
    // global reads coalesced (64 consecutive columns per k-row), ds_store_b128.
    {
        float4* sW4 = (float4*)sW;
        for (int q = tid; q < 64 * 64; q += 256) {   // 64 k-quads x 64 cols
            int kq = q >> 6;          // k-quad index: k = 4*kq
            int nl = q & 63;
            float4 v;
            v.x = W[(4 * kq + 0) * DDIM + n0 + nl];
            v.y = W[(4 * kq + 1) * DDIM + n0 + nl];
            v.z = W[(4 * kq + 2) * DDIM + n0 + nl];
            v.w = W[(4 * kq + 3) * DDIM + n0 + nl];
            sW4[kq * 64 + nl] = v;    // word kq*256 + nl*4 + {0,1,2,3}
        }
    }
    __syncthreads();

    const int wave  = tid >> 5;
    const int lane  = tid & 31;
    const int mtile = blockIdx.x * 8 + wave;
    if (mtile >= MTILES) return;           // wave-uniform: EXEC stays all-1s

    const int mbase = mtile * 16;
    const int mrow  = mbase + (lane & 15); // A fragment: M = lane%16
    const int kb    = (lane >> 4) << 1;    // K sub-offset: 0 (lanes 0-15) / 2 (lanes 16-31)
    const int ncol  = lane & 15;           // B/C fragment: N = lane%16

    v8f acc[4];
#pragma unroll
    for (int ns = 0; ns < 4; ++ns) {
        float bv = bias[n0 + ns * 16 + ncol];
        acc[ns] = (v8f){bv, bv, bv, bv, bv, bv, bv, bv};
    }

    // per k-step: one 8B global load for A, four aligned 8B LDS loads for B, 4 WMMAs
    const float2* arow = (const float2*)(A + (size_t)mrow * DDIM + kb);
    const float2* sW2  = (const float2*)sW;
    const int     bofs = (ncol * 2) + (kb >> 1);   // within-quad b64 offset for this lane
    for (int k = 0; k < DDIM; k += 4) {
        float2 af = arow[k >> 1];          // A[mrow, k+kb], A[mrow, k+kb+1]
        float a0 = af.x, a1 = af.y;
        if (RELU) { a0 = fmaxf(a0, 0.0f); a1 = fmaxf(a1, 0.0f); }
        v2f av; av.x = a0; av.y = a1;
        const float2* bq = sW2 + (k >> 2) * 128 + bofs;
#pragma unroll
        for (int ns = 0; ns < 4; ++ns) {
            float2 bf = bq[ns * 32];       // ds_load_b64, even-aligned pair
            v2f bv; bv.x = bf.x; bv.y = bf.y;
            acc[ns] = __builtin_amdgcn_wmma_f32_16x16x4_f32(
                false, av, false, bv, (short)0, acc[ns], false, false);
        }
    }

    // C/D layout: VGPR r, lane L -> M = r + (L/16)*8, N = L%16
    const int rbase = (lane >> 4) << 3;
#pragma unroll
    for (int ns = 0; ns < 4; ++ns) {
#pragma unroll
        for (int r = 0; r < 8; ++r) {
            H[(size_t)(mbase + rbase + r) * DDIM + n0 + ns * 16 + ncol] = acc[ns][r];
        }
    }
}

// ---------------------------------------------------------------- self term: Y = H * invdeg
__global__ void k_selfinit(const float* __restrict__ H, const float* __restrict__ invdeg,
                           float* __restrict__ Y) {
    int i = blockIdx.x;
    int f = threadIdx.x;
    Y[(size_t)i * DDIM + f] = H[(size_t)i * DDIM + f] * invdeg[i];
}

// ---------------------------------------------------------------- edge aggregation
// one wave per edge: gather h[src] (256 f32, 8 per lane, coalesced float4 loads),
// scale by dinv[src]*ew*dinv[dst], atomic-add into Y[dst].
__global__ __launch_bounds__(256) void k_edge_agg(const int* __restrict__ src,
                                                  const int* __restrict__ dst,
                                                  const float* __restrict__ ew,
                                                  const float* __restrict__ dinv,
                                                  const float* __restrict__ H,
                                                  float* __restrict__ Y) {
    const int lane = threadIdx.x & 31;
    const int e = blockIdx.x * 8 + (threadIdx.x >> 5);
    if (e >= NEDGES) return;               // wave-uniform
    const int s = src[e];
    const int d = dst[e];
    const float w = dinv[s] * ew[e] * dinv[d];

    const float4* hs = (const float4*)(H + (size_t)s * DDIM) + lane * 2;
    float4 v0 = hs[0];
    float4 v1 = hs[1];

    float* yd = Y + (size_t)d * DDIM + lane * 8;
    __hip_atomic_fetch_add(yd + 0, v0.x * w, __ATOMIC_RELAXED, __HIP_MEMORY_SCOPE_AGENT);
    __hip_atomic_fetch_add(yd + 1, v0.y * w, __ATOMIC_RELAXED, __HIP_MEMORY_SCOPE_AGENT);
    __hip_atomic_fetch_add(yd + 2, v0.z * w, __ATOMIC_RELAXED, __HIP_MEMORY_SCOPE_AGENT);
    __hip_atomic_fetch_add(yd + 3, v0.w * w, __ATOMIC_RELAXED, __HIP_MEMORY_SCOPE_AGENT);
    __hip_atomic_fetch_add(yd + 4, v1.x * w, __ATOMIC_RELAXED, __HIP_MEMORY_SCOPE_AGENT);
    __hip_atomic_fetch_add(yd + 5, v1.y * w, __ATOMIC_RELAXED, __HIP_MEMORY_SCOPE_AGENT);
    __hip_atomic_fetch_add(yd + 6, v1.z * w, __ATOMIC_RELAXED, __HIP_MEMORY_SCOPE_AGENT);
    __hip_atomic_fetch_add(yd + 7, v1.w * w, __ATOMIC_RELAXED, __HIP_MEMORY_SCOPE_AGENT);
}

// ---------------------------------------------------------------- launch
extern "C" void kernel_launch(void* const* d_in, const int* in_sizes, int n_in,
                              void* d_out, int out_size, void* d_ws, size_t ws_size,
                              hipStream_t stream) {
    (void)in_sizes; (void)n_in; (void)out_size; (void)ws_size;
    const float* x   = (const float*)d_in[0];
    const int*   ei  = (const int*)d_in[1];
    const float* ea  = (const float*)d_in[2];
    const float* W1  = (const float*)d_in[3];
    const float* b1  = (const float*)d_in[4];
    const float* W2  = (const float*)d_in[5];
    const float* b2  = (const float*)d_in[6];
    const float* W3  = (const float*)d_in[7];
    const float* b3  = (const float*)d_in[8];
    const float* mw1 = (const float*)d_in[9];
    const float* mb1 = (const float*)d_in[10];
    const float* mw2 = (const float*)d_in[11];
    const float* mb2 = (const float*)d_in[12];
    const int* src = ei;             // edge_index[0, :]
    const int* dst = ei + NEDGES;    // edge_index[1, :]

    float* ws     = (float*)d_ws;
    float* ew     = ws;                         // E floats
    float* deg    = ew     + NEDGES;
    float* dinv   = deg    + 50048;             // 64-mult padding keeps alignment
    float* invdeg = dinv   + 50048;
    float* Hb     = invdeg + 50048;             // 50000*256
    float* Yb     = Hb     + (size_t)NNODES * DDIM;
    float* out    = (float*)d_out;

    k_edge_mlp<<<(NEDGES + 255) / 256, 256, 0, stream>>>(ea, mw1, mb1, mw2, mb2, ew);
    k_deg_init<<<(NNODES + 255) / 256, 256, 0, stream>>>(deg);
    k_deg_acc<<<(NEDGES + 255) / 256, 256, 0, stream>>>(dst, ew, deg);
    k_dinv<<<(NNODES + 255) / 256, 256, 0, stream>>>(deg, dinv, invdeg);

    dim3 ggrid((MTILES + 7) / 8, DDIM / 64);   // 391 x 4
    dim3 agrid((NEDGES + 7) / 8);              // 100000

    // Layer 1: H = x@W1+b1 ; Y1 = self + agg
    k_gemm<false><<<ggrid, 256, 0, stream>>>(x, W1, b1, Hb);
    k_selfinit<<<NNODES, 256, 0, stream>>>(Hb, invdeg, Yb);
    k_edge_agg<<<agrid, 256, 0, stream>>>(src, dst, ew, dinv, Hb, Yb);

    // Layer 2: H = relu(Y1)@W2+b2 (relu fused into A-load, compile-time)
    k_gemm<true><<<ggrid, 256, 0, stream>>>(Yb, W2, b2, Hb);
    k_selfinit<<<NNODES, 256, 0, stream>>>(Hb, invdeg, Yb);
    k_edge_agg<<<agrid, 256, 0, stream>>>(src, dst, ew, dinv, Hb, Yb);

    // Layer 3: H = relu(Y2)@W3+b3 ; output straight to d_out (no final relu)
    k_gemm<true><<<ggrid, 256, 0, stream>>>(Yb, W3, b3, Hb);
    k_selfinit<<<NNODES, 256, 0, stream>>>(Hb, invdeg, out);
    k_edge_agg<<<agrid, 256, 0, stream>>>(src, dst, ew, dinv, Hb, out);
}